// SmoothLoss_73744588472820
// MI455X (gfx1250) — compile-verified
//
#include <hip/hip_runtime.h>

typedef float v2f __attribute__((ext_vector_type(2)));
typedef float v4f __attribute__((ext_vector_type(4)));
typedef float v8f __attribute__((ext_vector_type(8)));

#define BDIM 64
#define LDIM 1000000
#define NVEC (LDIM / 4)      // 250000 float4 per row
#define Q    20

// ---------------------------------------------------------------------------
// Kernel 0: zero the per-(row,bin) count table in workspace.
// ---------------------------------------------------------------------------
__global__ void SmoothLoss_zero_kernel(unsigned int* __restrict__ counts) {
    const int i = blockIdx.x * blockDim.x + threadIdx.x;
    if (i < BDIM * Q) counts[i] = 0u;
}

// ---------------------------------------------------------------------------
// Kernel 1: streaming histogram. Bandwidth-bound: 256 MB read once.
//  - b128 non-temporal loads (stream > L2, touched once)
//  - wave32 ballot/popcount counting into wave-uniform (SGPR) accumulators
//  - per-block LDS merge, then 20 global integer atomics per block
// ---------------------------------------------------------------------------
__global__ __launch_bounds__(256) void SmoothLoss_hist_kernel(
        const float* __restrict__ x, unsigned int* __restrict__ counts) {
    const int b = blockIdx.y;                       // row index 0..63
    const v4f* row = (const v4f*)(x + (size_t)b * LDIM);

    __shared__ unsigned int lhist[Q];
    if (threadIdx.x < Q) lhist[threadIdx.x] = 0u;
    __syncthreads();

    unsigned int cnt[Q];
#pragma unroll
    for (int q = 0; q < Q; ++q) cnt[q] = 0u;

    const unsigned int stride = gridDim.x * blockDim.x;
    // Block-uniform trip count: every lane of every wave stays converged so
    // ballots are wave-complete; lane 0's accumulators are exact totals.
    for (unsigned int base = blockIdx.x * blockDim.x; base < NVEC; base += stride) {
        const unsigned int i = base + threadIdx.x;
        const bool valid = (i < NVEC);
        const v4f v = __builtin_nontemporal_load(&row[valid ? i : 0u]);
        // bins: floor(x*20), x in [0,1) -> 0..19 (clamp for safety)
        int b0 = min((int)(v.x * 20.0f), Q - 1);
        int b1 = min((int)(v.y * 20.0f), Q - 1);
        int b2 = min((int)(v.z * 20.0f), Q - 1);
        int b3 = min((int)(v.w * 20.0f), Q - 1);
        if (!valid) { b0 = 255; b1 = 255; b2 = 255; b3 = 255; }
#pragma unroll
        for (int q = 0; q < Q; ++q) {
            cnt[q] += (unsigned int)__popcll(__ballot(b0 == q))
                    + (unsigned int)__popcll(__ballot(b1 == q))
                    + (unsigned int)__popcll(__ballot(b2 == q))
                    + (unsigned int)__popcll(__ballot(b3 == q));
        }
    }

    // One lane per wave flushes the wave-uniform counters into LDS.
    if ((threadIdx.x & 31) == 0) {
#pragma unroll
        for (int q = 0; q < Q; ++q) atomicAdd(&lhist[q], cnt[q]);
    }
    __syncthreads();
    if (threadIdx.x < Q)
        atomicAdd(&counts[b * Q + threadIdx.x], lhist[threadIdx.x]);
}

// ---------------------------------------------------------------------------
// Kernel 2 (single wave32): compute 1280 entropy terms and sum them with 20
// accumulating V_WMMA_F32_16X16X4_F32 ops (B = ones, C accumulates). Since B
// is all-ones and we sum over every (m,k) slot, the result is exactly the sum
// of all 1280 values independent of A's lane layout.
// ---------------------------------------------------------------------------
__global__ __launch_bounds__(32) void SmoothLoss_entropy_wmma_kernel(
        const unsigned int* __restrict__ counts, float* __restrict__ out) {
    const int lane = threadIdx.x;     // 0..31 (wave32, EXEC all ones)
    const int m    = lane & 15;
    const int kh   = lane >> 4;
    const float invL = 1.0f / (float)LDIM;

    v8f acc = {0.f, 0.f, 0.f, 0.f, 0.f, 0.f, 0.f, 0.f};
    v2f bones; bones.x = 1.0f; bones.y = 1.0f;

    for (int t = 0; t < Q; ++t) {
        v2f a;
#pragma unroll
        for (int j = 0; j < 2; ++j) {
            const int idx = t * 64 + m * 4 + kh * 2 + j;   // bijective over 0..1279
            const unsigned int c = counts[idx];
            const float p = (float)c * invL;
            a[j] = (c != 0u) ? (-p * __log2f(p)) : 0.0f;   // branchless cndmask
        }
        // 8 args: (neg_a, A, neg_b, B, c_mod, C, reuse_a, reuse_b)
        acc = __builtin_amdgcn_wmma_f32_16x16x4_f32(
            false, a, false, bones, (short)0, acc, false, false);
    }

    // Each lane sums its 8 D registers: lanes 0-15 hold sum of rows 0-7,
    // lanes 16-31 hold sum of rows 8-15. Combine halves with one shuffle.
    float s = acc[0] + acc[1] + acc[2] + acc[3] + acc[4] + acc[5] + acc[6] + acc[7];
    s += __shfl_xor(s, 16, 32);
    if (lane == 0) out[0] = s;
}

// ---------------------------------------------------------------------------
extern "C" void kernel_launch(void* const* d_in, const int* in_sizes, int n_in,
                              void* d_out, int out_size, void* d_ws, size_t ws_size,
                              hipStream_t stream) {
    const float* x = (const float*)d_in[0];          // [64, 1000000] f32
    unsigned int* counts = (unsigned int*)d_ws;      // [64, 20] u32
    float* out = (float*)d_out;                      // scalar f32

    SmoothLoss_zero_kernel<<<(BDIM * Q + 255) / 256, 256, 0, stream>>>(counts);

    dim3 grid(64, BDIM);                             // 4096 blocks x 8 waves
    SmoothLoss_hist_kernel<<<grid, 256, 0, stream>>>(x, counts);

    SmoothLoss_entropy_wmma_kernel<<<1, 32, 0, stream>>>(counts, out);
}